// Decoder_6966436954585
// MI455X (gfx1250) — compile-verified
//
#include <hip/hip_runtime.h>
#include <hip/hip_bf16.h>

#define NF       64
#define PC       192          // perception channels = 3*NF
#define IMGD     128
#define IMGN     16384        // 128*128
#define BATCH    16
#define NCALLS   16
#define LATD     512
#define STATE_ELEMS (BATCH * NF * IMGN)          // 16,777,216 per step slot
#define CLAMP_ELEMS (BATCH * 3 * IMGN)           // 786,432
#define EMB_ELEMS   ((NCALLS + 1) * STATE_ELEMS) // 285,212,672

typedef __attribute__((ext_vector_type(2))) float v2f;
typedef __attribute__((ext_vector_type(8))) float v8f;
typedef __attribute__((ext_vector_type(4))) unsigned int u32x4;
typedef __attribute__((ext_vector_type(8))) int i32x8;
typedef __attribute__((ext_vector_type(4))) int i32x4;

#if __has_builtin(__builtin_amdgcn_tensor_load_to_lds) && __has_builtin(__builtin_amdgcn_s_wait_tensorcnt)
#define USE_TDM 1
#else
#define USE_TDM 0
#endif

// ---------- branchless zero-padded Sobel (both gradients from 8 shared taps) ----------
// Addresses are clamped in-range (always-valid loads); zero padding applied as
// 0/1 multiplicative masks -> no EXEC divergence, no branches.
__device__ __forceinline__ void sobel_both(const float* __restrict__ base,
                                           int hu, int hm, int hd,
                                           int wl, int wc, int wr,
                                           float mu, float md, float ml, float mr,
                                           float& gx, float& gy) {
    const float* __restrict__ ru = base + (hu << 7);
    const float* __restrict__ rm = base + (hm << 7);
    const float* __restrict__ rd = base + (hd << 7);
    const float ul = ru[wl], uc = ru[wc], ur = ru[wr];
    const float l  = rm[wl],             r  = rm[wr];
    const float dl = rd[wl], dc = rd[wc], dr = rd[wr];
    const float uL = ml * ul, uR = mr * ur;
    const float dL = ml * dl, dR = mr * dr;
    gx = fmaf(mu, uL - uR, fmaf(md, dL - dR, 2.0f * fmaf(ml, l, -(mr * r))));
    gy = mu * fmaf(2.0f, uc, uL + uR) - md * fmaf(2.0f, dc, dL + dR);
}

__device__ __forceinline__ float wave_reduce(float v) {
    #pragma unroll
    for (int off = 16; off > 0; off >>= 1) v += __shfl_xor(v, off, 32);
    return v;
}

// ---------- kernel 1: dynamic weights  out[g] = dot(lat[b], Wt[j]) + bvec[j] ----------
__global__ __launch_bounds__(256) void dot_kernel(const float* __restrict__ lat,
                                                  const float* __restrict__ Wt,
                                                  const float* __restrict__ bvec,
                                                  float* __restrict__ out, int rows) {
    const int g = blockIdx.x * 8 + (threadIdx.x >> 5);
    const int lane = threadIdx.x & 31;
    const int b = g / rows;
    const int j = g - b * rows;
    const float* __restrict__ lrow = lat + (size_t)b * LATD;
    const float* __restrict__ wrow = Wt + (size_t)j * LATD;
    float s = 0.0f;
    #pragma unroll 4
    for (int k = lane; k < LATD; k += 32) s = fmaf(lrow[k], wrow[k], s);
    s = wave_reduce(s);
    if (lane == 0) out[g] = s + bvec[j];
}

// ---------- kernel 2: state0 = zeros with spike at (c=0, h=64, w=64) ----------
__global__ __launch_bounds__(256) void init_kernel(float* __restrict__ embs0) {
    const int idx = blockIdx.x * 256 + threadIdx.x;            // < STATE_ELEMS
    embs0[idx] = ((idx & (NF * IMGN - 1)) == (64 * 128 + 64)) ? 1.0f : 0.0f;
}

// ---------- kernel 3: per-(b,c) stats of x, gx, gy over H*W ----------
__global__ __launch_bounds__(256) void stats_kernel(const float* __restrict__ state,
                                                    float* __restrict__ meanA,
                                                    float* __restrict__ rstdA) {
    const int c = blockIdx.x;        // 0..63 (source channel)
    const int b = blockIdx.y;        // 0..15
    const float* __restrict__ Sb = state + ((size_t)b * NF + c) * IMGN;

    float s[6] = {0, 0, 0, 0, 0, 0}; // x, x^2, gx, gx^2, gy, gy^2
    for (int p = threadIdx.x; p < IMGN; p += 256) {
        const int h = p >> 7, w = p & 127;
        const int hu = (h > 0) ? h - 1 : 0, hd = (h < 127) ? h + 1 : 127;
        const int wl = (w > 0) ? w - 1 : 0, wr = (w < 127) ? w + 1 : 127;
        const float mu = (h > 0) ? 1.0f : 0.0f, md = (h < 127) ? 1.0f : 0.0f;
        const float ml = (w > 0) ? 1.0f : 0.0f, mr = (w < 127) ? 1.0f : 0.0f;
        const float x  = Sb[p];
        float gx, gy;
        sobel_both(Sb, hu, h, hd, wl, w, wr, mu, md, ml, mr, gx, gy);
        s[0] += x;  s[1] = fmaf(x, x, s[1]);
        s[2] += gx; s[3] = fmaf(gx, gx, s[3]);
        s[4] += gy; s[5] = fmaf(gy, gy, s[5]);
    }

    __shared__ float red[8 * 6];
    const int wave = threadIdx.x >> 5, lane = threadIdx.x & 31;
    #pragma unroll
    for (int i = 0; i < 6; ++i) {
        const float r = wave_reduce(s[i]);
        if (lane == 0) red[wave * 6 + i] = r;
    }
    __syncthreads();
    if (threadIdx.x == 0) {
        float tot[6] = {0, 0, 0, 0, 0, 0};
        for (int wv = 0; wv < 8; ++wv)
            #pragma unroll
            for (int i = 0; i < 6; ++i) tot[i] += red[wv * 6 + i];
        const float inv = 1.0f / (float)IMGN;
        #pragma unroll
        for (int i = 0; i < 3; ++i) {
            const float m = tot[2 * i] * inv;
            const float v = fmaxf(tot[2 * i + 1] * inv - m * m, 0.0f);
            meanA[b * PC + i * NF + c] = m;
            rstdA[b * PC + i * NF + c] = rsqrtf(v + 1e-5f);
        }
    }
}

// ---------- kernel 4: WMMA dyna-conv + leaky residual ----------
// grid (128, BATCH), 256 threads = 8 waves; each workgroup covers one image
// row (128 px); each wave owns one 16-pixel tile x all 64 output channels.
__global__ __launch_bounds__(256) void step_matmul(const float* __restrict__ state,
                                                   float* __restrict__ next,
                                                   const float* __restrict__ Wall,
                                                   const float* __restrict__ ball,
                                                   const float* __restrict__ meanA,
                                                   const float* __restrict__ rstdA,
                                                   const float* __restrict__ leak) {
    __shared__ float wlds[NF * PC];                 // 48 KB: W[b] row-major [o][i]
    __shared__ float mean_s[PC], rstd_s[PC], bias_s[NF];

    const int b = blockIdx.y;
    const int tid = threadIdx.x;

    const float* __restrict__ Wb = Wall + (size_t)b * NF * PC;

#if USE_TDM
    // Tensor Data Mover: one wave DMAs the whole contiguous 48KB W[b] tile
    // into LDS (D# per CDNA5 ISA 8.3/8.4: count=1, type=2(image),
    // data_size=4B, 1-row tensor, tile_dim0 = NF*PC elements, no cluster).
    if (tid < 32) {
        const unsigned long long ga = (unsigned long long)(uintptr_t)Wb;
        const unsigned int ldsa = (unsigned int)(uintptr_t)&wlds[0];
        u32x4 g0;
        g0[0] = 1u;                                                  // count=1, user desc
        g0[1] = ldsa;                                                // lds_addr [63:32]
        g0[2] = (unsigned int)(ga & 0xffffffffu);                    // global_addr [95:64]
        g0[3] = (unsigned int)((ga >> 32) & 0x01ffffffu) | (2u << 30); // ga[56:32] | type=2
        i32x8 g1;
        const unsigned int NE = (unsigned int)(NF * PC);             // 12288 elements
        g1[0] = (int)(2u << 16);                                     // data_size=4B; mask=0
        g1[1] = (int)((NE & 0xffffu) << 16);                         // tensor_dim0[15:0] @bits63:48
        g1[2] = (int)((NE >> 16) | (1u << 16));                      // tensor_dim0[31:16] | tensor_dim1=1
        g1[3] = (int)(NE << 16);                                     // tile_dim0 @bits127:112
        g1[4] = 1;                                                   // tile_dim1=1, tile_dim2=0
        g1[5] = (int)NE;                                             // tensor_dim0_stride[31:0]
        g1[6] = (int)((NE & 0xffffu) << 16);                         // dim0_stride[47:32]=0 | dim1_stride[15:0]
        g1[7] = (int)(NE >> 16);                                     // dim1_stride[47:16]
        i32x4 gz4 = {0, 0, 0, 0};
        i32x8 gz8 = {0, 0, 0, 0, 0, 0, 0, 0};
        // 6-arg form (clang-23 / therock-10.0 headers)
        __builtin_amdgcn_tensor_load_to_lds(g0, g1, gz4, gz4, gz8, 0);
        __builtin_amdgcn_s_wait_tensorcnt((short)0);
    }
#else
    for (int i = tid; i < NF * PC; i += 256) wlds[i] = Wb[i];
#endif
    for (int i = tid; i < PC; i += 256) { mean_s[i] = meanA[b * PC + i]; rstd_s[i] = rstdA[b * PC + i]; }
    for (int i = tid; i < NF; i += 256) bias_s[i] = ball[b * NF + i];
    __syncthreads();

    const float lf = fminf(fmaxf(leak[0], 0.001f), 1000.0f);

    const int lane = tid & 31;
    const int n    = lane & 15;        // N (pixel) within tile
    const int hi   = lane >> 4;        // half-wave -> K pair select
    const int ptile = blockIdx.x * 8 + (tid >> 5);
    const int p = ptile * 16 + n;
    const int h = p >> 7, w = p & 127;

    // clamped neighbor coords + 0/1 padding masks (loop-invariant per lane)
    const int hu = (h > 0) ? h - 1 : 0, hd = (h < 127) ? h + 1 : 127;
    const int wl = (w > 0) ? w - 1 : 0, wr = (w < 127) ? w + 1 : 127;
    const float mu = (h > 0) ? 1.0f : 0.0f, md = (h < 127) ? 1.0f : 0.0f;
    const float ml = (w > 0) ? 1.0f : 0.0f, mr = (w < 127) ? 1.0f : 0.0f;

    const float* __restrict__ Sb = state + (size_t)b * NF * IMGN;

    v8f acc[4] = {v8f{}, v8f{}, v8f{}, v8f{}};

    // c0 = 4*kt + 2*hi is this lane's K (reg0 = c0, reg1 = c0+1).
    #define WMMA_STEP(c0, b0, b1)                                               \
    do {                                                                        \
        v2f bf;                                                                 \
        bf.x = ((b0) - mean_s[(c0)]) * rstd_s[(c0)];                            \
        bf.y = ((b1) - mean_s[(c0) + 1]) * rstd_s[(c0) + 1];                    \
        const v2f a0 = *(const v2f*)&wlds[(0  + n) * PC + (c0)];                \
        const v2f a1 = *(const v2f*)&wlds[(16 + n) * PC + (c0)];                \
        const v2f a2 = *(const v2f*)&wlds[(32 + n) * PC + (c0)];                \
        const v2f a3 = *(const v2f*)&wlds[(48 + n) * PC + (c0)];                \
        acc[0] = __builtin_amdgcn_wmma_f32_16x16x4_f32(false, a0, false, bf, (short)0, acc[0], false, false); \
        acc[1] = __builtin_amdgcn_wmma_f32_16x16x4_f32(false, a1, false, bf, (short)0, acc[1], false, false); \
        acc[2] = __builtin_amdgcn_wmma_f32_16x16x4_f32(false, a2, false, bf, (short)0, acc[2], false, false); \
        acc[3] = __builtin_amdgcn_wmma_f32_16x16x4_f32(false, a3, false, bf, (short)0, acc[3], false, false); \
    } while (0)

    // ---- K tiles 0..15: raw x channels ----
    for (int kt = 0; kt < 16; ++kt) {
        const int c0 = kt * 4 + hi * 2;
        const float b0 = Sb[(size_t)c0 * IMGN + p];
        const float b1 = Sb[(size_t)(c0 + 1) * IMGN + p];
        WMMA_STEP(c0, b0, b1);
    }
    // ---- K tiles 16..47: sobel-x (64+cc) and sobel-y (128+cc) share 8 taps
    //      per source channel cc, so compute both gradients per load set. ----
    for (int kt2 = 0; kt2 < 16; ++kt2) {
        const int cc0 = kt2 * 4 + hi * 2;                 // source channel pair
        const float* __restrict__ base0 = Sb + (size_t)cc0 * IMGN;
        float gx0, gy0, gx1, gy1;
        sobel_both(base0,        hu, h, hd, wl, w, wr, mu, md, ml, mr, gx0, gy0);
        sobel_both(base0 + IMGN, hu, h, hd, wl, w, wr, mu, md, ml, mr, gx1, gy1);
        WMMA_STEP(64  + cc0, gx0, gx1);
        WMMA_STEP(128 + cc0, gy0, gy1);
    }
    #undef WMMA_STEP

    // D layout: reg r holds M = 16*m + 8*hi + r, N = lane&15
    float* __restrict__ Nb = next + (size_t)b * NF * IMGN;
    #pragma unroll
    for (int m = 0; m < 4; ++m) {
        #pragma unroll
        for (int r = 0; r < 8; ++r) {
            const int o = m * 16 + hi * 8 + r;
            const float v = fmaf(lf, acc[m][r] + bias_s[o], Sb[o * IMGN + p]);
            Nb[o * IMGN + p] = v;
        }
    }
}

// ---------- kernel 5: raw + clamp of first 3 channels ----------
__global__ __launch_bounds__(256) void final_kernel(const float* __restrict__ emb16,
                                                    float* __restrict__ clamped,
                                                    float* __restrict__ raw) {
    const int idx = blockIdx.x * 256 + threadIdx.x;    // < CLAMP_ELEMS
    const int b = idx / (3 * IMGN);
    const int rem = idx - b * 3 * IMGN;                // c*IMGN + p, c < 3
    const float v = emb16[(size_t)b * NF * IMGN + rem];
    raw[idx] = v;
    clamped[idx] = fminf(fmaxf(v, -1.0f), 1.0f);
}

// ---------- host ----------
extern "C" void kernel_launch(void* const* d_in, const int* in_sizes, int n_in,
                              void* d_out, int out_size, void* d_ws, size_t ws_size,
                              hipStream_t stream) {
    const float* lat  = (const float*)d_in[0];
    const float* leak = (const float*)d_in[1];
    const float* dww  = (const float*)d_in[2];
    const float* dwb  = (const float*)d_in[3];
    const float* dbw  = (const float*)d_in[4];
    const float* dbb  = (const float*)d_in[5];
    float* out = (float*)d_out;
    float* ws  = (float*)d_ws;

    // workspace layout (floats): W 196608 | bias 1024 | mean 3072 | rstd 3072  (~815 KB)
    float* Wws   = ws;
    float* bws   = ws + 196608;
    float* meanA = ws + 197632;
    float* rstdA = ws + 200704;

    float* clampedP = out;
    float* embs     = out + CLAMP_ELEMS;
    float* rawP     = out + (size_t)CLAMP_ELEMS + (size_t)EMB_ELEMS;

    // dynamic weights: W (16 x 12288) and bias (16 x 64)
    dot_kernel<<<dim3((BATCH * 12288) / 8), dim3(256), 0, stream>>>(lat, dww, dwb, Wws, 12288);
    dot_kernel<<<dim3((BATCH * 64) / 8),    dim3(256), 0, stream>>>(lat, dbw, dbb, bws, 64);

    // state0 into out_embs[0]
    init_kernel<<<dim3(STATE_ELEMS / 256), dim3(256), 0, stream>>>(embs);

    for (int t = 0; t < NCALLS; ++t) {
        const float* st = embs + (size_t)t * STATE_ELEMS;
        float* nx       = embs + (size_t)(t + 1) * STATE_ELEMS;
        stats_kernel<<<dim3(NF, BATCH), dim3(256), 0, stream>>>(st, meanA, rstdA);
        step_matmul<<<dim3(128, BATCH), dim3(256), 0, stream>>>(st, nx, Wws, bws, meanA, rstdA, leak);
    }

    final_kernel<<<dim3(CLAMP_ELEMS / 256), dim3(256), 0, stream>>>(
        embs + (size_t)NCALLS * STATE_ELEMS, clampedP, rawP);
}